// LocalPosEnc2D_59837484367920
// MI455X (gfx1250) — compile-verified
//
#include <hip/hip_runtime.h>
#include <hip/hip_fp16.h>

// ---- problem constants (match reference) ----
#define GRID_N   512
#define GRID_NV  513
#define OUT_DIM  24      // D0(8) + 4*N_FREQ(16)
#define EPS_F    1e-6f
#define BLOCK    256

typedef unsigned int u32x4 __attribute__((ext_vector_type(4)));

union Row48 {            // one 24-channel fp16 grid row = 48 bytes
    u32x4   q[3];
    __half2 h[12];
};

__global__ __launch_bounds__(BLOCK, 1) void
posenc2d_kernel(const float2* __restrict__ coords,
                const __half* __restrict__ grids,
                __half*       __restrict__ out,
                int B)
{
    __shared__ float2 sc[BLOCK];

    const int tid = threadIdx.x;
    const int pid = (int)blockIdx.x * BLOCK + tid;          // B <= 2^21, fits i32
    const int pclamp = pid < B ? pid : B - 1;

    // ---- CDNA5 async path: stage this block's coord tile into LDS ----
    // (per-lane 8B global->LDS DMA, tracked by ASYNCcnt)
    {
        unsigned long long gsrc = (unsigned long long)(const void*)(coords + pclamp);
        unsigned lds = (unsigned)(unsigned long long)(const void*)&sc[tid];
        asm volatile("global_load_async_to_lds_b64 %0, %1, off"
                     :
                     : "v"(lds), "v"(gsrc)
                     : "memory");
        asm volatile("s_wait_asynccnt 0x0" ::: "memory");
    }
    __syncthreads();

    if (pid >= B) return;

    float u = sc[tid].x;
    float v = sc[tid].y;
    u = fminf(fmaxf(u, 0.0f), 1.0f - EPS_F);   // -> v_med3_num_f32
    v = fminf(fmaxf(v, 0.0f), 1.0f - EPS_F);

    const float fu = u * (float)GRID_N;
    const float fv = v * (float)GRID_N;
    const int   iu = (int)fu;                  // clamp guarantees 0..511
    const int   iv = (int)fv;
    const float lu = fu - (float)iu;
    const float lv = fv - (float)iv;

    const int idx00 = iu + iv * GRID_NV;

    const u32x4* p00 = (const u32x4*)(grids + idx00 * OUT_DIM);
    const u32x4* p10 = (const u32x4*)(grids + (idx00 + 1) * OUT_DIM);
    const u32x4* p01 = (const u32x4*)(grids + (idx00 + GRID_NV) * OUT_DIM);
    const u32x4* p11 = (const u32x4*)(grids + (idx00 + GRID_NV + 1) * OUT_DIM);

    // issue all 12 b128 loads up-front (grid is L2-resident; hide L2 latency)
    Row48 r00, r10, r01, r11;
    r00.q[0] = p00[0]; r00.q[1] = p00[1]; r00.q[2] = p00[2];
    r10.q[0] = p10[0]; r10.q[1] = p10[1]; r10.q[2] = p10[2];
    r01.q[0] = p01[0]; r01.q[1] = p01[1]; r01.q[2] = p01[2];
    r11.q[0] = p11[0]; r11.q[1] = p11[1]; r11.q[2] = p11[2];

    // bilinear weights, broadcast as packed fp16 -> v_pk_fma_f16 path
    const float om_lu = 1.0f - lu, om_lv = 1.0f - lv;
    const __half2 w00 = __float2half2_rn(om_lu * om_lv);
    const __half2 w10 = __float2half2_rn(lu * om_lv);
    const __half2 w01 = __float2half2_rn(om_lu * lv);
    const __half2 w11 = __float2half2_rn(lu * lv);

    Row48 o;
    #pragma unroll
    for (int i = 0; i < 12; ++i) {
        __half2 acc = __hmul2(r00.h[i], w00);
        acc = __hfma2(r10.h[i], w10, acc);
        acc = __hfma2(r01.h[i], w01, acc);
        acc = __hfma2(r11.h[i], w11, acc);
        o.h[i] = acc;
    }

    // v_sin_f32 / v_cos_f32 compute sin/cos(2*pi*x): feed revolutions
    // directly -> sin(2*pi * 2^k * lu) with no range reduction.
    float cu[4], su[4], cv[4], sv[4];
    #pragma unroll
    for (int k = 0; k < 4; ++k) {
        const float s  = (float)(1 << k);
        const float tu = lu * s, tv = lv * s;
        cu[k] = __builtin_amdgcn_cosf(tu);
        su[k] = __builtin_amdgcn_sinf(tu);
        cv[k] = __builtin_amdgcn_cosf(tv);
        sv[k] = __builtin_amdgcn_sinf(tv);
    }

    // pe = [cos(uu)x4, sin(uu)x4, cos(vv)x4, sin(vv)x4], packed to half2
    __half2 pe2[8];
    pe2[0] = __floats2half2_rn(cu[0], cu[1]);
    pe2[1] = __floats2half2_rn(cu[2], cu[3]);
    pe2[2] = __floats2half2_rn(su[0], su[1]);
    pe2[3] = __floats2half2_rn(su[2], su[3]);
    pe2[4] = __floats2half2_rn(cv[0], cv[1]);
    pe2[5] = __floats2half2_rn(cv[2], cv[3]);
    pe2[6] = __floats2half2_rn(sv[0], sv[1]);
    pe2[7] = __floats2half2_rn(sv[2], sv[3]);

    // gate channels 8..23 (= half2 slots 4..11)
    #pragma unroll
    for (int i = 0; i < 8; ++i)
        o.h[4 + i] = __hmul2(o.h[4 + i], pe2[i]);

    // Output is a 100 MB write-once stream: non-temporal stores keep the
    // 12.6 MB grid resident in L2 instead of letting the output churn it.
    u32x4* po = (u32x4*)(out + pid * OUT_DIM);
    __builtin_nontemporal_store(o.q[0], po + 0);
    __builtin_nontemporal_store(o.q[1], po + 1);
    __builtin_nontemporal_store(o.q[2], po + 2);
}

extern "C" void kernel_launch(void* const* d_in, const int* in_sizes, int n_in,
                              void* d_out, int out_size, void* d_ws, size_t ws_size,
                              hipStream_t stream)
{
    const float2* coords = (const float2*)d_in[0];
    const __half* grids  = (const __half*)d_in[1];
    __half*       out    = (__half*)d_out;

    const int B = in_sizes[0] / 2;               // coords is (B, 2) f32
    const int nblocks = (B + BLOCK - 1) / BLOCK; // 8192 blocks for B = 2M

    posenc2d_kernel<<<nblocks, BLOCK, 0, stream>>>(coords, grids, out, B);
}